// Decoder_35029753266434
// MI455X (gfx1250) — compile-verified
//
#include <hip/hip_runtime.h>

typedef __attribute__((ext_vector_type(2))) float v2f;
typedef __attribute__((ext_vector_type(8))) float v8f;
typedef __attribute__((ext_vector_type(4))) int   v4i;

#define NB 32
#define NA 16
#define NN 10000
#define NE 256
#define NH 256
#define NO 2
#define TW 5                         // n-tiles per wave in phase 1 (625 = 5*125)

static_assert(NB == 32, "phase 2 maps lane==batch, needs B==wave32");

#if defined(__has_builtin)
#if __has_builtin(__builtin_amdgcn_global_load_async_to_lds_b128)
#define HAVE_ASYNC_LDS 1
#endif
#endif

// ---------------------------------------------------------------------------
// Phase 1: P[n,h] = emb[n,:] . W1[1:,h] + b1[h]   (batch-independent GEMM)
// One wave32 per (5 n-tiles x 1 h-tile): per K-step load one B fragment and
// five A fragments, issue five V_WMMA_F32_16X16X4_F32 -> B reuse x5.
// A frag (16x4 f32): lanes 0-15 hold K=0,1; lanes 16-31 hold K=2,3; M=lane&15
// B frag (4x16 f32): same K split; N=lane&15
// C/D (16x16 f32): VGPR v -> M=v (+8 for upper half), N=lane&15
// ---------------------------------------------------------------------------
__global__ __launch_bounds__(256) void decoder_phase1_wmma(
    const float* __restrict__ emb, const float* __restrict__ W1,
    const float* __restrict__ b1, float* __restrict__ P) {
  const int wave = threadIdx.x >> 5;
  const int lane = threadIdx.x & 31;
  const int tile = blockIdx.x * 8 + wave;   // 0..1999 (125 n-groups x 16 h-tiles)
  const int tg   = tile >> 4;               // 0..124  n-group of 5 tiles
  const int th   = tile & 15;               // 0..15
  const int m0   = tg * (16 * TW);          // first row of the 80-row strip
  const int h0   = th << 4;
  const int half = lane >> 4;
  const int lm   = lane & 15;

  const float* ea = emb + (size_t)(m0 + lm) * NE + 2 * half;
  const float* wb = W1 + (size_t)(1 + 2 * half) * NH + h0 + lm;

  v8f acc[TW];
  #pragma unroll
  for (int t = 0; t < TW; ++t) acc[t] = (v8f){0.f,0.f,0.f,0.f,0.f,0.f,0.f,0.f};

  #pragma unroll 2
  for (int k = 0; k < NE; k += 4) {
    v2f bf;
    bf.x = wb[0];
    bf.y = wb[NH];
    wb += 4 * NH;
    #pragma unroll
    for (int t = 0; t < TW; ++t) {
      v2f a = *(const v2f*)(ea + k + (size_t)t * 16 * NE);
      acc[t] = __builtin_amdgcn_wmma_f32_16x16x4_f32(
          false, a, false, bf, (short)0, acc[t], false, false);
    }
  }

  const float bias = b1[h0 + lm];
  #pragma unroll
  for (int t = 0; t < TW; ++t) {
    float* p = P + (size_t)(m0 + t * 16 + 8 * half) * NH + h0 + lm;
    #pragma unroll
    for (int v = 0; v < 8; ++v) p[(size_t)v * NH] = acc[t][v] + bias;
  }
}

// ---------------------------------------------------------------------------
// Phase 2: out[b,n,:] = relu(P[n,:] + assigned[b,n]*W1[0,:]) . W2 + b2
// lane == batch (B==32==wave32). One wave per n; P row / W1 row0 / W2 are
// lane-uniform -> broadcast ds_load_b128 from LDS, zero cross-lane reduction.
// P row staged via async global->LDS copy when available (ASYNCcnt path).
// ---------------------------------------------------------------------------
__global__ __launch_bounds__(256) void decoder_phase2(
    const float* __restrict__ abs_actions, const int* __restrict__ assign,
    const float* __restrict__ W1, const float* __restrict__ W2,
    const float* __restrict__ b2, const float* __restrict__ P,
    float* __restrict__ out) {
  __shared__ float  s_w1[NH];        // W1 row 0                    (1 KB)
  __shared__ float2 s_w2[NH];        // W2 rows as float2           (2 KB)
  __shared__ float  s_p[8][NH];      // one P row per wave          (8 KB)

  const int tid  = threadIdx.x;
  const int wave = tid >> 5;
  const int lane = tid & 31;
  const int n    = blockIdx.x * 8 + wave;   // 0..9999, uniform per wave

  // block-wide staging of the shared weights
  s_w1[tid] = W1[tid];
  s_w2[tid] = ((const float2*)W2)[tid];

  // stage this wave's P row (1 KB): 2 x b128 per lane
  const float* prow_g = P + (size_t)n * NH;
#if HAVE_ASYNC_LDS
  {
    v4i* gsrc = (v4i*)prow_g;            // generic pointers, per builtin proto
    v4i* ldst = (v4i*)&s_p[wave][0];
    __builtin_amdgcn_global_load_async_to_lds_b128(gsrc + lane, ldst + lane, 0, 0);
    __builtin_amdgcn_global_load_async_to_lds_b128(gsrc + 32 + lane,
                                                   ldst + 32 + lane, 0, 0);
#if __has_builtin(__builtin_amdgcn_s_wait_asynccnt)
    __builtin_amdgcn_s_wait_asynccnt(0);
#else
    asm volatile("s_wait_asynccnt 0x0" ::: "memory");
#endif
  }
#else
  {
    float4 a = ((const float4*)prow_g)[lane];
    float4 c = ((const float4*)prow_g)[lane + 32];
    ((float4*)&s_p[wave][0])[lane]      = a;
    ((float4*)&s_p[wave][0])[lane + 32] = c;
  }
#endif

  // per-lane (== per-batch) assigned scalar
  const int b   = lane;
  const float s = abs_actions[b * NA + assign[(size_t)b * NN + n]];

  __syncthreads();

  float acc0 = 0.f, acc1 = 0.f;
  #pragma unroll 4
  for (int h = 0; h < NH; h += 4) {
    float4 pv  = *(const float4*)&s_p[wave][h];   // broadcast ds_load_b128
    float4 wv  = *(const float4*)&s_w1[h];
    float2 w20 = s_w2[h + 0];
    float2 w21 = s_w2[h + 1];
    float2 w22 = s_w2[h + 2];
    float2 w23 = s_w2[h + 3];
    float hv;
    hv = fmaf(s, wv.x, pv.x); hv = fmaxf(hv, 0.f);
    acc0 = fmaf(hv, w20.x, acc0); acc1 = fmaf(hv, w20.y, acc1);
    hv = fmaf(s, wv.y, pv.y); hv = fmaxf(hv, 0.f);
    acc0 = fmaf(hv, w21.x, acc0); acc1 = fmaf(hv, w21.y, acc1);
    hv = fmaf(s, wv.z, pv.z); hv = fmaxf(hv, 0.f);
    acc0 = fmaf(hv, w22.x, acc0); acc1 = fmaf(hv, w22.y, acc1);
    hv = fmaf(s, wv.w, pv.w); hv = fmaxf(hv, 0.f);
    acc0 = fmaf(hv, w23.x, acc0); acc1 = fmaf(hv, w23.y, acc1);
  }

  float2 r;
  r.x = acc0 + b2[0];
  r.y = acc1 + b2[1];
  *(float2*)(out + ((size_t)b * NN + n) * NO) = r;
}

extern "C" void kernel_launch(void* const* d_in, const int* in_sizes, int n_in,
                              void* d_out, int out_size, void* d_ws, size_t ws_size,
                              hipStream_t stream) {
  (void)in_sizes; (void)n_in; (void)out_size; (void)ws_size;
  const float* abs_actions = (const float*)d_in[0];   // (B, A)
  const int*   assignments = (const int*)d_in[1];     // (B, N)
  const float* emb         = (const float*)d_in[2];   // (N, E)
  const float* W1          = (const float*)d_in[3];   // (1+E, H)
  const float* b1          = (const float*)d_in[4];   // (H,)
  const float* W2          = (const float*)d_in[5];   // (H, OUT)
  const float* b2          = (const float*)d_in[6];   // (OUT,)

  float* P   = (float*)d_ws;     // (N, H) = 10.24 MB scratch, L2-resident
  float* out = (float*)d_out;    // (B, N, OUT) f32

  // Phase 1: 125 n-groups x 16 h-tiles = 2000 waves -> 250 blocks of 8 waves
  decoder_phase1_wmma<<<(NN / (16 * TW)) * (NH / 16) / 8, 256, 0, stream>>>(
      emb, W1, b1, P);
  // Phase 2: one wave per n -> 1250 blocks of 8 waves
  decoder_phase2<<<NN / 8, 256, 0, stream>>>(abs_actions, assignments,
                                             W1, W2, b2, P, out);
}